// ShortestPathBias_1726576853584
// MI455X (gfx1250) — compile-verified
//
#include <hip/hip_runtime.h>
#include <hip/hip_bf16.h>
#include <stdint.h>

#define NUM_GRAPHS 256
#define NN 128                // nodes per graph
#define DEG 16
#define EPG (NN * DEG)        // edges per graph = 2048
#define NUM_EDGES (NUM_GRAPHS * EPG)
#define NUM_HEADS 8
#define SCALE_F 10.0f

#define THREADS 1024          // 32 wave32 waves per block, one block per graph

__global__ __launch_bounds__(THREADS)
void spb_fw_kernel(const int* __restrict__ edge_index,    // [2, E] flat: src row then dst row
                   const float* __restrict__ edge_weight, // [E]
                   float* __restrict__ out)               // [G*H, n, n]
{
    __shared__ float d[NN * NN];        // 64 KB distance tile (later reused as bias tile)
    __shared__ float redm[32];
    __shared__ int   redi[32];
    __shared__ float s_unreach, s_coef;

    const int g   = blockIdx.x;
    const int tid = threadIdx.x;
    const float INFF = __builtin_inff();

    // ---- 1) init: diag 0, off-diag +inf ----
    for (int idx = tid; idx < NN * NN; idx += THREADS) {
        int i = idx >> 7, j = idx & (NN - 1);
        d[idx] = (i == j) ? 0.0f : INFF;
    }
    __syncthreads();

    // ---- 2) scatter edges with LDS atomic-min (uint trick: all values >= 0) ----
    {
        const int* srcp = edge_index + (size_t)g * EPG;
        const int* dstp = edge_index + (size_t)NUM_EDGES + (size_t)g * EPG;
        const float* wp = edge_weight + (size_t)g * EPG;
        const int base = g * NN;
        for (int e = tid; e < EPG; e += THREADS) {
            int s = srcp[e], t = dstp[e];
            if (s != t) {                        // self-loops masked to +inf in reference
                int u = s - base, v = t - base;
                float w = wp[e];
                atomicMin((unsigned int*)&d[u * NN + v], __float_as_uint(w));
            }
        }
    }
    __syncthreads();

    // ---- 3) Floyd-Warshall, register-blocked ----
    // thread -> (row r0, columns [cb, cb+16)); 8 threads per row.
    // The 16 owned values live in VGPRs b0..b3 across the whole k-loop.
    // LDS freshness protocol per pivot k (readers need row k and column k):
    //   end of pivot k: row-(k+1) threads flush all 4 quads; every other row's
    //   owner of the quad containing column k+1 flushes that quad. All
    //   overlapping concurrent writes are same-value (pivot k leaves row k /
    //   column k unchanged because the diagonal is 0), hence benign.
    const int r0  = tid >> 3;
    const int cb  = (tid & 7) << 4;
    const int myq = tid & 7;                       // which 16-col block of the row
    float4* rowi4 = (float4*)&d[r0 * NN + cb];

    float4 b0 = rowi4[0], b1 = rowi4[1], b2 = rowi4[2], b3 = rowi4[3];

    for (int k = 0; k < NN; ++k) {
        float dik = d[r0 * NN + k];                         // fresh per protocol
        const float4* rowk4 = (const float4*)&d[k * NN + cb]; // fresh per protocol
        float4 a0 = rowk4[0], a1 = rowk4[1], a2 = rowk4[2], a3 = rowk4[3];

#define FW_UPD(B, A) \
        B.x = fminf(B.x, dik + A.x); \
        B.y = fminf(B.y, dik + A.y); \
        B.z = fminf(B.z, dik + A.z); \
        B.w = fminf(B.w, dik + A.w);
        FW_UPD(b0, a0) FW_UPD(b1, a1) FW_UPD(b2, a2) FW_UPD(b3, a3)
#undef FW_UPD

        const int kn = k + 1;
        if (kn < NN) {
            if (r0 == kn) {
                // next pivot row: flush all 4 quads
                rowi4[0] = b0; rowi4[1] = b1; rowi4[2] = b2; rowi4[3] = b3;
            } else if ((kn >> 4) == myq) {
                // owner of the quad containing column k+1: flush that quad
                const int q = (kn >> 2) & 3;
                float4 pub = (q == 0) ? b0 : (q == 1) ? b1 : (q == 2) ? b2 : b3;
                ((float4*)&d[r0 * NN])[kn >> 2] = pub;
            }
        }
        __syncthreads();
    }

    // ---- 4) reductions straight from registers (wave32 shuffles) ----
    {
        float m = 0.0f;                  // diag zeros guarantee finite max >= 0
        int anyinf = 0;
#define FW_ACC(V) { float v_ = (V); if (v_ < INFF) m = fmaxf(m, v_); else anyinf = 1; }
        FW_ACC(b0.x) FW_ACC(b0.y) FW_ACC(b0.z) FW_ACC(b0.w)
        FW_ACC(b1.x) FW_ACC(b1.y) FW_ACC(b1.z) FW_ACC(b1.w)
        FW_ACC(b2.x) FW_ACC(b2.y) FW_ACC(b2.z) FW_ACC(b2.w)
        FW_ACC(b3.x) FW_ACC(b3.y) FW_ACC(b3.z) FW_ACC(b3.w)
#undef FW_ACC
#pragma unroll
        for (int o = 16; o > 0; o >>= 1) {
            m = fmaxf(m, __shfl_xor(m, o));
            anyinf |= __shfl_xor(anyinf, o);
        }
        const int wid = tid >> 5, lane = tid & 31;
        if (lane == 0) { redm[wid] = m; redi[wid] = anyinf; }
        __syncthreads();
        if (wid == 0) {
            float mm = redm[lane];
            int   ai = redi[lane];
#pragma unroll
            for (int o = 16; o > 0; o >>= 1) {
                mm = fmaxf(mm, __shfl_xor(mm, o));
                ai |= __shfl_xor(ai, o);
            }
            if (lane == 0) {
                float fmax = mm;                        // finite, >= 0
                float unreach = 2.0f * fmax;
                float maxval = fmaxf(ai ? unreach : fmax, 1e-8f);
                s_unreach = unreach;
                s_coef    = -SCALE_F / maxval;          // bias = coef * dist
            }
        }
    }
    __syncthreads();

    // ---- 5) bias transform in registers, single flush of the tile to LDS ----
    {
        const float unreach = s_unreach;
        const float coef    = s_coef;
#define FW_BIAS(V) { float v_ = (V); (V) = coef * ((v_ < INFF) ? v_ : unreach); }
        FW_BIAS(b0.x) FW_BIAS(b0.y) FW_BIAS(b0.z) FW_BIAS(b0.w)
        FW_BIAS(b1.x) FW_BIAS(b1.y) FW_BIAS(b1.z) FW_BIAS(b1.w)
        FW_BIAS(b2.x) FW_BIAS(b2.y) FW_BIAS(b2.z) FW_BIAS(b2.w)
        FW_BIAS(b3.x) FW_BIAS(b3.y) FW_BIAS(b3.z) FW_BIAS(b3.w)
#undef FW_BIAS
        rowi4[0] = b0; rowi4[1] = b1; rowi4[2] = b2; rowi4[3] = b3;
    }
    __syncthreads();

    // ---- 6) stream 8 head-broadcast copies out of LDS with async stores ----
    // Tile = 16384 floats = 4096 x b128 chunks. Each thread moves chunks
    // (tid + q*1024), q=0..3, for each of the 8 heads: 32 async b128 stores.
    {
        const uint64_t outBase =
            (uint64_t)(uintptr_t)out +
            (uint64_t)g * NUM_HEADS * NN * NN * sizeof(float);
        const uint32_t ldsBase = (uint32_t)(uintptr_t)&d[0];  // low 32 bits = LDS byte offset

#pragma unroll
        for (int h = 0; h < NUM_HEADS; ++h) {
            const uint64_t headBase = outBase + (uint64_t)h * NN * NN * sizeof(float);
#pragma unroll
            for (int q = 0; q < 4; ++q) {
                const uint32_t byteOff = (uint32_t)(tid + q * THREADS) * 16u;
                const uint64_t gaddr = headBase + byteOff;
                const uint32_t laddr = ldsBase + byteOff;
                asm volatile("global_store_async_from_lds_b128 %0, %1, off"
                             :
                             : "v"(gaddr), "v"(laddr)
                             : "memory");
            }
        }
        asm volatile("s_wait_asynccnt 0" ::: "memory");
    }
}

extern "C" void kernel_launch(void* const* d_in, const int* in_sizes, int n_in,
                              void* d_out, int out_size, void* d_ws, size_t ws_size,
                              hipStream_t stream) {
    (void)in_sizes; (void)n_in; (void)d_ws; (void)ws_size; (void)out_size;
    const int*   edge_index  = (const int*)d_in[0];    // [2, E] int32
    const float* edge_weight = (const float*)d_in[1];  // [E] f32
    // d_in[2] (batch), d_in[3]/d_in[4] (scalars) are implied by the fixed
    // contiguous-block layout (graph g owns nodes [g*128, (g+1)*128)).
    float* out = (float*)d_out;                        // [G*H, 128, 128] f32

    dim3 grid(NUM_GRAPHS);
    dim3 block(THREADS);
    spb_fw_kernel<<<grid, block, 0, stream>>>(edge_index, edge_weight, out);
}